// RBFImageCollection_180388627074
// MI455X (gfx1250) — compile-verified
//
#include <hip/hip_runtime.h>

typedef __attribute__((ext_vector_type(2))) float v2f;
typedef __attribute__((ext_vector_type(8))) float v8f;

#define IMGS 32
#define RBFS 128
#define NPTS 16384                    // 128*128 grid points
#define WAVES_PER_BLOCK 8
#define TPB (WAVES_PER_BLOCK * 32)    // 256 threads = 8 wave32
#define PTS_PER_BLOCK (WAVES_PER_BLOCK * 16)   // 128 points per block
#define BLOCKS_PER_IMG (NPTS / PTS_PER_BLOCK)  // 128

// q[r,p] = F_r . G_p with
//   G_p = [gx^2, gy^2, gx*gy, gx, gy, 1, 0, 0]            (K padded 6->8)
//   F_r = kexp * [a, b, 2c, -2(a*cx+c*cy), -2(b*cy+c*cx), a*cx^2+b*cy^2+2c*cx*cy, 0, 0]
// with kexp = -0.5*log2(e) pre-folded into F, so the WMMA accumulator is
// directly log2(exp(-q/2)) and the inner loop is pure v_exp_f32 + v_fmac_f32.
// Two chained V_WMMA_F32_16X16X4_F32 per 16x16 (rbf x point) tile.
__global__ __launch_bounds__(TPB)
void RBFImageCollection_wmma_kernel(const float* __restrict__ centers,
                                    const float* __restrict__ covs,
                                    const float* __restrict__ amps,
                                    const float* __restrict__ grid,
                                    float* __restrict__ out)
{
    // F features padded to 8 floats/RBF so A-fragment loads are uniform b64s.
    __shared__ float Flds[RBFS * 8];
    __shared__ float Alds[RBFS];

    const int img  = blockIdx.x / BLOCKS_PER_IMG;
    const int pblk = blockIdx.x % BLOCKS_PER_IMG;
    const int tid  = threadIdx.x;

    // ---- one-time per-block: build scaled F features for this image's RBFs ----
    if (tid < RBFS) {
        const float kexp = -0.72134752044448170f;  // -0.5 * log2(e)
        const int r    = tid;
        const int base = img * RBFS + r;
        const float cx = centers[base * 2 + 0];
        const float cy = centers[base * 2 + 1];
        const float a  = covs[base * 3 + 0];
        const float b  = covs[base * 3 + 1];
        const float c  = covs[base * 3 + 2];
        Flds[r * 8 + 0] = kexp * a;
        Flds[r * 8 + 1] = kexp * b;
        Flds[r * 8 + 2] = kexp * (2.0f * c);
        Flds[r * 8 + 3] = kexp * (-2.0f * (a * cx + c * cy));
        Flds[r * 8 + 4] = kexp * (-2.0f * (b * cy + c * cx));
        Flds[r * 8 + 5] = kexp * (a * cx * cx + b * cy * cy + 2.0f * c * cx * cy);
        Flds[r * 8 + 6] = 0.0f;   // K-pad
        Flds[r * 8 + 7] = 0.0f;   // K-pad
        Alds[r] = amps[base];
    }
    __syncthreads();

    const int  wave = tid >> 5;
    const int  lane = tid & 31;
    const int  n    = lane & 15;        // point (B/N) or rbf-row (A/M) index in tile
    const bool hihalf = lane >= 16;     // lanes 16-31 carry K=2,3 (wmma1) / K=6,7 (wmma2)
    const int  koff = hihalf ? 2 : 0;   // lane-half K offset within the 8-float record
    const int  p    = pblk * PTS_PER_BLOCK + wave * 16 + n;

    const float gx = grid[p * 2 + 0];
    const float gy = grid[p * 2 + 1];

    // ---- B fragments (point features), invariant across R-tiles ----
    // B[k][n]: vgpr = k&1, lane half selects k<2 vs k>=2.
    v2f b01, b23;
    b01.x = hihalf ? (gx * gy) : (gx * gx);   // K=2 : K=0
    b01.y = hihalf ? gx        : (gy * gy);   // K=3 : K=1
    b23.x = hihalf ? 0.0f      : gy;          // K=6 : K=4
    b23.y = hihalf ? 0.0f      : 1.0f;        // K=7 : K=5

    float s = 0.0f;

#pragma unroll
    for (int rt = 0; rt < RBFS / 16; ++rt) {
        // ---- A fragments: one ds_load_2addr_b64 worth of uniform loads ----
        const float* Fr = &Flds[(rt * 16 + n) * 8 + koff];
        const v2f a01 = *(const v2f*)(Fr);        // lo: K0,K1   hi: K2,K3
        const v2f a23 = *(const v2f*)(Fr + 4);    // lo: K4,K5   hi: K6,K7 (zeros)

        v8f acc = {};
        // (neg_a, A, neg_b, B, c_mod, C, reuse_a, reuse_b)
        acc = __builtin_amdgcn_wmma_f32_16x16x4_f32(false, a01, false, b01,
                                                    (short)0, acc, false, false);
        acc = __builtin_amdgcn_wmma_f32_16x16x4_f32(false, a23, false, b23,
                                                    (short)0, acc, false, false);

        // acc[j] = log2(exp(-q/2)) for rbf (rt*16 + j + (hihalf?8:0)), point p.
        const float* Ar = &Alds[rt * 16 + (hihalf ? 8 : 0)];
#pragma unroll
        for (int j = 0; j < 8; ++j) {
            s += Ar[j] * __builtin_amdgcn_exp2f(acc[j]);
        }
    }

    // lanes L and L^16 hold the two halves (r 0-7 vs 8-15 of each tile) of point p
    s += __shfl_xor(s, 16, 32);

    // sigmoid(z) = 1 / (1 + 2^(-z*log2 e))
    const float sig = __builtin_amdgcn_rcpf(
        1.0f + __builtin_amdgcn_exp2f(-1.4426950408889634f * s));

    if (!hihalf) {
        out[img * NPTS + p] = sig;
    }
}

extern "C" void kernel_launch(void* const* d_in, const int* in_sizes, int n_in,
                              void* d_out, int out_size, void* d_ws, size_t ws_size,
                              hipStream_t stream) {
    const float* centers = (const float*)d_in[0];
    const float* covs    = (const float*)d_in[1];
    const float* amps    = (const float*)d_in[2];
    const float* grid    = (const float*)d_in[3];
    float*       out     = (float*)d_out;

    dim3 grd(IMGS * BLOCKS_PER_IMG);   // 4096 blocks
    dim3 blk(TPB);                     // 256 threads (8 wave32)
    RBFImageCollection_wmma_kernel<<<grd, blk, 0, stream>>>(centers, covs, amps, grid, out);
}